// DWTForward3d_13477607375543
// MI455X (gfx1250) — compile-verified
//
#include <hip/hip_runtime.h>
#include <cstdint>
#include <cstddef>

typedef float v4f __attribute__((ext_vector_type(4)));

#define HAAR_SCALE 0.35355339059327373f   // 2^-1.5 = (1/sqrt(2))^3

// ---- CDNA5 async global->LDS copy (tracked by ASYNCcnt) -------------------
// One lane moves 16B from its global address to its LDS byte address.
__device__ __forceinline__ void async_copy_b128(uint32_t lds_addr, const float* gaddr) {
    asm volatile("global_load_async_to_lds_b128 %0, %1, off"
                 :: "v"(lds_addr), "v"(gaddr)
                 : "memory");
}
__device__ __forceinline__ void wait_async_le8() {
    asm volatile("s_wait_asynccnt 8" ::: "memory");
}
__device__ __forceinline__ void wait_async_0() {
    asm volatile("s_wait_asynccnt 0" ::: "memory");
}

// One level of 3D Haar DWT on NC cubes of edge inW = 8<<wqBits.
// Each thread computes 4 consecutive W outputs (one 2x2x8-float input brick),
// staged to LDS by async b128 copies, double-buffered across iterations.
// Subband 0 (LLL) -> llOut[nc][outVol]; subbands 1..7 -> hiOut[nc*7+b-1][outVol].
__global__ __launch_bounds__(256) void haar3d_level_kernel(
    const float* __restrict__ in, float* __restrict__ llOut,
    float* __restrict__ hiOut, int wqBits, int hBits, int niter)
{
    // 2 buffers x 256 threads x 8 chunks of 16B = 64 KB
    __shared__ v4f smem[2 * 256 * 8];

    const int      tid  = threadIdx.x;
    const uint32_t T    = gridDim.x * blockDim.x;        // threads per iteration
    const uint32_t g0   = blockIdx.x * blockDim.x + tid; // work id, iter 0

    const uint32_t wqMask = (1u << wqBits) - 1u;
    const uint32_t hMask  = (1u << hBits) - 1u;
    const int      inW    = 8 << wqBits;
    const uint32_t inRow  = (uint32_t)inW;
    const uint32_t inSlab = (uint32_t)inW * inW;
    const uint32_t inVol  = inSlab * inW;
    const int      outW   = 4 << wqBits;
    const uint32_t outRow  = (uint32_t)outW;
    const uint32_t outSlab = (uint32_t)outW * outW;
    const uint32_t outVol  = outSlab * outW;

    // 32-bit LDS byte addresses of this thread's two staging buffers.
    uint32_t ldsBase[2];
    ldsBase[0] = (uint32_t)(uintptr_t)&smem[(size_t)tid * 8];
    ldsBase[1] = (uint32_t)(uintptr_t)&smem[2048 + (size_t)tid * 8];

    auto issue = [&](uint32_t g, int buf) {
        const uint32_t wq = g & wqMask;
        const uint32_t h  = (g >> wqBits) & hMask;
        const uint32_t d  = (g >> (wqBits + hBits)) & hMask;
        const uint32_t nc = g >> (wqBits + 2 * hBits);
        const float* base = in + (size_t)nc * inVol
                               + (size_t)(2u * d) * inSlab
                               + (size_t)(2u * h) * inRow
                               + (size_t)wq * 8u;
        const uint32_t lb = ldsBase[buf];
        #pragma unroll
        for (int i = 0; i < 2; ++i)
            #pragma unroll
            for (int j = 0; j < 2; ++j) {
                const int r = i * 2 + j;
                const float* p = base + (size_t)i * inSlab + (size_t)j * inRow;
                async_copy_b128(lb + (uint32_t)r * 32u,       p);
                async_copy_b128(lb + (uint32_t)r * 32u + 16u, p + 4);
            }
    };

    int buf = 0;
    issue(g0, 0);

    for (int it = 0; it < niter; ++it) {
        const uint32_t g = g0 + (uint32_t)it * T;

        if (it + 1 < niter) {          // prefetch next brick into other buffer
            issue(g0 + (uint32_t)(it + 1) * T, buf ^ 1);
            wait_async_le8();          // 16 outstanding -> drain the first 8
        } else {
            wait_async_0();
        }

        // ---- pull staged brick out of LDS: p[r=2i+j][0..7] -----------------
        const v4f* ch = &smem[(buf ? 2048 : 0) + (size_t)tid * 8];
        float p[4][8];
        #pragma unroll
        for (int r = 0; r < 4; ++r) {
            v4f a = ch[2 * r], b = ch[2 * r + 1];
            p[r][0] = a.x; p[r][1] = a.y; p[r][2] = a.z; p[r][3] = a.w;
            p[r][4] = b.x; p[r][5] = b.y; p[r][6] = b.z; p[r][7] = b.w;
        }

        // ---- separable Haar butterflies, 4 outputs along W ----------------
        float out[8][4];               // [subband b = 4fw+2fh+fd][o]
        #pragma unroll
        for (int o = 0; o < 4; ++o) {
            float s1l[4], s1h[4];      // stage 1: along W (k)
            #pragma unroll
            for (int r = 0; r < 4; ++r) {
                const float x0 = p[r][2 * o], x1 = p[r][2 * o + 1];
                s1l[r] = x0 + x1;
                s1h[r] = x0 - x1;
            }
            float s2[2][2][2];         // stage 2: along H (j); [i][fw][fh]
            #pragma unroll
            for (int i = 0; i < 2; ++i) {
                s2[i][0][0] = s1l[2 * i] + s1l[2 * i + 1];
                s2[i][0][1] = s1l[2 * i] - s1l[2 * i + 1];
                s2[i][1][0] = s1h[2 * i] + s1h[2 * i + 1];
                s2[i][1][1] = s1h[2 * i] - s1h[2 * i + 1];
            }
            #pragma unroll
            for (int fw = 0; fw < 2; ++fw)
                #pragma unroll
                for (int fh = 0; fh < 2; ++fh) {
                    const float lo = s2[0][fw][fh] + s2[1][fw][fh];
                    const float hi = s2[0][fw][fh] - s2[1][fw][fh];
                    out[4 * fw + 2 * fh + 0][o] = lo * HAAR_SCALE;
                    out[4 * fw + 2 * fh + 1][o] = hi * HAAR_SCALE;
                }
        }

        // ---- stores -------------------------------------------------------
        const uint32_t wq = g & wqMask;
        const uint32_t h  = (g >> wqBits) & hMask;
        const uint32_t d  = (g >> (wqBits + hBits)) & hMask;
        const uint32_t nc = g >> (wqBits + 2 * hBits);
        const size_t sp = (size_t)d * outSlab + (size_t)h * outRow + (size_t)wq * 4u;

        // LL subband: regular (RT) store — re-read by the next level (L2-resident).
        v4f vll = { out[0][0], out[0][1], out[0][2], out[0][3] };
        *(v4f*)(llOut + (size_t)nc * outVol + sp) = vll;

        // High subbands: streamed once, never re-read -> non-temporal stores.
        #pragma unroll
        for (int b = 1; b < 8; ++b) {
            v4f vv = { out[b][0], out[b][1], out[b][2], out[b][3] };
            __builtin_nontemporal_store(
                vv, (v4f*)(hiOut + ((size_t)nc * 7u + (size_t)(b - 1)) * outVol + sp));
        }

        buf ^= 1;
    }
}

extern "C" void kernel_launch(void* const* d_in, const int* in_sizes, int n_in,
                              void* d_out, int out_size, void* d_ws, size_t ws_size,
                              hipStream_t stream) {
    (void)in_sizes; (void)n_in; (void)out_size; (void)ws_size;
    const float* x   = (const float*)d_in[0];   // (2,4,128,128,128) f32
    float*       out = (float*)d_out;

    // d_out layout (flat, return order): ll2 | highs1 | highs2
    const size_t LL2  = 2ull * 4 * 32 * 32 * 32;          //   262144
    const size_t HI1  = 2ull * 4 * 7 * 64 * 64 * 64;      // 14680064
    float* ll2 = out;
    float* hi1 = out + LL2;
    float* hi2 = out + LL2 + HI1;
    float* ll1 = (float*)d_ws;                            // 8 MB scratch (2,4,64^3)

    // Level 1: 8 cubes of 128^3 -> 64^3. 8<<16 = 524288 work items,
    // 512 blocks x 256 threads x 4 iterations (double-buffered async pipeline).
    haar3d_level_kernel<<<512, 256, 0, stream>>>(x, ll1, hi1, /*wqBits=*/4, /*hBits=*/6, /*niter=*/4);

    // Level 2: 8 cubes of 64^3 -> 32^3. 8<<13 = 65536 work items, one pass.
    haar3d_level_kernel<<<256, 256, 0, stream>>>(ll1, ll2, hi2, /*wqBits=*/3, /*hBits=*/5, /*niter=*/1);
}